// Filter_module_9019431322046
// MI455X (gfx1250) — compile-verified
//
#include <hip/hip_runtime.h>
#include <cstdint>
#include <cstddef>

// ---------------------------------------------------------------------------
// Four fixed 3x3 stencils + L2 norm over the 4 responses, per single-channel
// 512x512 map (96 maps total). HBM-bound: ~200 MB traffic, ~8.6 us floor.
// CDNA5 path: async global->LDS staging (global_load_async_to_lds_b128,
// ASYNCcnt, s_wait_asynccnt) of an 18-row halo tile, 9-point neighborhood
// compute out of LDS, raw v_sqrt_f32, non-temporal b128 output stores.
// ---------------------------------------------------------------------------

#define MAP_W   512
#define MAP_H   512
#define N_MAPS  96          // B*C = 32*3
#define TILE_H  16
#define LROWS   (TILE_H + 2)
#define PITCH   520         // floats per LDS row; interior starts at index 4 (16B aligned)

typedef float float4_t __attribute__((ext_vector_type(4)));
typedef int   v4i      __attribute__((vector_size(16)));   // matches builtin pointee type

#if defined(__gfx1250__) && __has_builtin(__builtin_amdgcn_global_load_async_to_lds_b128) && \
    __has_builtin(__builtin_amdgcn_s_wait_asynccnt)
#define USE_ASYNC_LDS 1
#else
#define USE_ASYNC_LDS 0
#endif

#if __has_builtin(__builtin_amdgcn_sqrtf)
#define FAST_SQRT(x) __builtin_amdgcn_sqrtf(x)   // raw v_sqrt_f32, ~1 ulp
#else
#define FAST_SQRT(x) __builtin_sqrtf(x)
#endif

__global__ __launch_bounds__(256) void stencil4_norm_kernel(const float* __restrict__ x,
                                                            float* __restrict__ out) {
    __shared__ float lds[LROWS * PITCH];

    const int tid  = threadIdx.x;        // 0..255
    const int tile = blockIdx.x;         // 0..31  (row tile within a map)
    const int map  = blockIdx.y;         // 0..95  (B*C maps)
    const int row0 = tile * TILE_H;      // first output row of this tile

    const float* __restrict__ xm = x   + (size_t)map * MAP_H * MAP_W;
    float* __restrict__       om = out + (size_t)map * MAP_H * MAP_W;

    // ---- zero halo columns (LDS float index 3 = col -1, index 516 = col 512) ----
    if (tid < 2 * LROWS) {
        const int r = tid >> 1;
        const int c = (tid & 1) ? 516 : 3;
        lds[r * PITCH + c] = 0.0f;
    }
    // ---- zero out-of-image halo rows (top / bottom of the map) ----
    if (row0 == 0) {
        for (int i = tid; i < 514; i += 256) lds[0 * PITCH + 3 + i] = 0.0f;
    }
    if (row0 + TILE_H == MAP_H) {
        for (int i = tid; i < 514; i += 256) lds[(LROWS - 1) * PITCH + 3 + i] = 0.0f;
    }

    // ---- stage rows (row0-1 .. row0+TILE_H) into LDS: 18 rows x 128 b128 chunks.
    //      Per iteration lrow is wave-uniform, lanes cover 32 consecutive 16B
    //      chunks -> fully coalesced 512B/wave async transfers.
    for (int q = tid; q < LROWS * (MAP_W / 4); q += 256) {
        const int lrow = q >> 7;              // 0..17
        const int c4   = q & 127;             // 0..127 -> float col 4*c4
        const int gr   = row0 - 1 + lrow;     // global row
        if (gr >= 0 && gr < MAP_H) {
            const float* gp = xm + (size_t)gr * MAP_W + c4 * 4;
            float*       lp = &lds[lrow * PITCH + 4 + c4 * 4];   // 16B-aligned LDS addr
#if USE_ASYNC_LDS
            __builtin_amdgcn_global_load_async_to_lds_b128(
                (__attribute__((address_space(1))) v4i*)gp,
                (__attribute__((address_space(3))) v4i*)lp,
                /*offset=*/0, /*cpol=*/0);
#else
            *(float4_t*)lp = *(const float4_t*)gp;
#endif
        }
    }
#if USE_ASYNC_LDS
    __builtin_amdgcn_s_wait_asynccnt(0);
#endif
    __syncthreads();

    // ---- compute: each thread does an 8-row x 4-col strip with row rolling ----
    const int c4  = tid & 127;           // column group
    const int col = c4 * 4;              // output cols col..col+3
    const int rg  = tid >> 7;            // 0 or 1
    const int tr0 = rg * 8;              // first tile-row handled by this thread
    const int base = 3 + col;            // LDS float index of col-1

    // LDS row (tr) holds global row row0+tr-1 -> neighborhood rows tr, tr+1, tr+2
    float r0[6], r1[6], r2[6];
#pragma unroll
    for (int j = 0; j < 6; ++j) r0[j] = lds[(tr0 + 0) * PITCH + base + j];
#pragma unroll
    for (int j = 0; j < 6; ++j) r1[j] = lds[(tr0 + 1) * PITCH + base + j];

#pragma unroll
    for (int tr = 0; tr < 8; ++tr) {
        const int t = tr0 + tr;
#pragma unroll
        for (int j = 0; j < 6; ++j) r2[j] = lds[(t + 2) * PITCH + base + j];

        float4_t o;
#pragma unroll
        for (int j = 0; j < 4; ++j) {
            const float c2 = -2.0f * r1[j + 1];
            const float g1 = (r1[j]     + r1[j + 2]) + c2;   // [1 -2 1] horizontal
            const float g2 = (r0[j + 1] + r2[j + 1]) + c2;   // vertical
            const float g3 = (r0[j + 2] + r2[j]    ) + c2;   // anti-diagonal
            const float g4 = (r0[j]     + r2[j + 2]) + c2;   // main diagonal
            float s = g1 * g1;
            s = __builtin_fmaf(g2, g2, s);
            s = __builtin_fmaf(g3, g3, s);
            s = __builtin_fmaf(g4, g4, s);
            o[j] = FAST_SQRT(s);
        }
        // Output is never re-read: non-temporal store keeps L2 free for the
        // inter-tile halo rows that neighboring blocks re-read.
        __builtin_nontemporal_store(o, (float4_t*)&om[(size_t)(row0 + t) * MAP_W + col]);

#pragma unroll
        for (int j = 0; j < 6; ++j) { r0[j] = r1[j]; r1[j] = r2[j]; }
    }
}

extern "C" void kernel_launch(void* const* d_in, const int* in_sizes, int n_in,
                              void* d_out, int out_size, void* d_ws, size_t ws_size,
                              hipStream_t stream) {
    (void)in_sizes; (void)n_in; (void)d_ws; (void)ws_size; (void)out_size;
    const float* x = (const float*)d_in[0];
    float* out = (float*)d_out;
    dim3 grid(MAP_H / TILE_H, N_MAPS);   // 32 x 96 = 3072 blocks
    stencil4_norm_kernel<<<grid, 256, 0, stream>>>(x, out);
}